// PairBilinear_83365315216073
// MI455X (gfx1250) — compile-verified
//
#include <hip/hip_runtime.h>

#define GG       64          // grid G
#define NPAIR    8           // pairs staged per block (8 * 32KB = 256KB LDS)
#define DIM      4096
#define BATCH    16384
#define NPAIRS   (DIM / 2)   // 2048
#define NTHREADS 256
#define BPB      2048        // batch rows per block

typedef __attribute__((ext_vector_type(4))) unsigned int u32x4;
typedef __attribute__((ext_vector_type(8))) int          i32x8;
typedef __attribute__((ext_vector_type(4))) int          i32x4;
typedef __attribute__((ext_vector_type(4))) float        f32x4;

__global__ __launch_bounds__(NTHREADS, 1)
void pair_bilinear_tdm_kernel(const float* __restrict__ x,
                              const float* __restrict__ Y,
                              float* __restrict__ out)
{
    extern __shared__ float smem[];   // NPAIR*2*GG*GG floats = 256 KB

    const int pg  = blockIdx.y;             // pair-group index [0, 256)
    const int p0  = pg * NPAIR;
    const int tid = threadIdx.x;
    const size_t ybase = (size_t)p0 * 2 * GG * GG;   // Y region is contiguous

#if __has_builtin(__builtin_amdgcn_tensor_load_to_lds)
    // ---- Tensor Data Mover: DMA the 8-pair grid block (256 KB) into LDS ----
    if (tid < 32) {   // wave 0 only issues the DMA (EXEC ignored by TDM)
        // LDS byte address: low 32 bits of the flat shared-aperture address
        unsigned lds_base = (unsigned)(unsigned long long)smem;
        unsigned long long gaddr = (unsigned long long)(Y + ybase);

        // D# group 0: count=1 | lds_addr | global_addr[56:0] | type=2
        u32x4 g0;
        g0.x = 1u;
        g0.y = lds_base;
        g0.z = (unsigned)gaddr;
        g0.w = ((unsigned)(gaddr >> 32) & 0x01FFFFFFu) | 0x80000000u;

        // 2-D copy: rows of 4096 f32 elements, 16 rows, stride 4096
        const unsigned row_elems = 4096;
        const unsigned rows = (NPAIR * 2 * GG * GG) / row_elems;   // 16
        i32x8 g1;
        g1.s0 = (int)(2u << 16);                                  // data_size = 4B
        g1.s1 = (int)(row_elems << 16);                           // tensor_dim0[15:0]
        g1.s2 = (int)((row_elems >> 16) | (rows << 16));          // tensor_dim0[31:16] | tensor_dim1[15:0]
        g1.s3 = (int)((rows >> 16) | (row_elems << 16));          // tensor_dim1[31:16] | tile_dim0
        g1.s4 = (int)rows;                                        // tile_dim1 (tile_dim2 = 0)
        g1.s5 = (int)row_elems;                                   // tensor_dim0_stride[31:0]
        g1.s6 = 0;
        g1.s7 = 0;
        i32x4 g2 = {0, 0, 0, 0};
        i32x4 g3 = {0, 0, 0, 0};
#if defined(__clang_major__) && (__clang_major__ >= 23)
        i32x8 gz = {0, 0, 0, 0, 0, 0, 0, 0};
        __builtin_amdgcn_tensor_load_to_lds(g0, g1, g2, g3, gz, 0);
#else
        __builtin_amdgcn_tensor_load_to_lds(g0, g1, g2, g3, 0);
#endif
        __builtin_amdgcn_s_wait_tensorcnt(0);
    }
#else
    // Fallback staging if the TDM builtin is unavailable
    {
        const f32x4* src = (const f32x4*)(Y + ybase);
        f32x4* dst = (f32x4*)smem;
        for (int i = tid; i < (NPAIR * 2 * GG * GG) / 4; i += NTHREADS)
            dst[i] = src[i];
    }
#endif
    __syncthreads();

    const float scale = (float)(GG - 1);
    const int rows_per_thread = BPB / NTHREADS;   // 8
    const int bbase = blockIdx.x * BPB;
    const int coff  = pg * (2 * NPAIR);           // 16-float column slice

#pragma unroll 1
    for (int r = 0; r < rows_per_thread; ++r) {
        const int b = bbase + r * NTHREADS + tid;
        const float* xrow = x + (size_t)b * DIM + coff;

        // Prefetch next iteration's x slice (global_prefetch_b8)
        if (r + 1 < rows_per_thread)
            __builtin_prefetch(xrow + (size_t)NTHREADS * DIM, 0, 1);

        // 64B-aligned 4x b128 NT loads of this row's 16 coordinates
        // (non-temporal: x is streaming, keep Y resident in the 192MB L2)
        float xa[2 * NPAIR];
#pragma unroll
        for (int q = 0; q < 4; ++q) {
            f32x4 t = __builtin_nontemporal_load((const f32x4*)xrow + q);
            xa[4 * q + 0] = t.x; xa[4 * q + 1] = t.y;
            xa[4 * q + 2] = t.z; xa[4 * q + 3] = t.w;
        }

        float res[2 * NPAIR];
#pragma unroll
        for (int k = 0; k < NPAIR; ++k) {
            float u = fminf(fmaxf(xa[2 * k + 0], 0.0f), 1.0f) * scale;
            float v = fminf(fmaxf(xa[2 * k + 1], 0.0f), 1.0f) * scale;
            int i0 = (int)floorf(u);
            int j0 = (int)floorf(v);
            i0 = i0 < 0 ? 0 : (i0 > GG - 2 ? GG - 2 : i0);
            j0 = j0 < 0 ? 0 : (j0 > GG - 2 ? GG - 2 : j0);
            float fu = u - (float)i0;
            float fv = v - (float)j0;
            float w00 = (1.0f - fu) * (1.0f - fv);
            float w01 = (1.0f - fu) * fv;
            float w10 = fu * (1.0f - fv);
            float w11 = fu * fv;

            const float* s0 = smem + k * (2 * GG * GG) + i0 * GG + j0;  // ch 0
            float a00 = s0[0], a01 = s0[1], a10 = s0[GG], a11 = s0[GG + 1];
            res[2 * k + 0] = w00 * a00 + w01 * a01 + w10 * a10 + w11 * a11;

            const float* s1 = s0 + GG * GG;                             // ch 1
            float b00 = s1[0], b01 = s1[1], b10 = s1[GG], b11 = s1[GG + 1];
            res[2 * k + 1] = w00 * b00 + w01 * b01 + w10 * b10 + w11 * b11;
        }

        // NT streaming stores of the 16 outputs
        float* orow = out + (size_t)b * DIM + coff;
#pragma unroll
        for (int q = 0; q < 4; ++q) {
            f32x4 v4;
            v4.x = res[4 * q + 0]; v4.y = res[4 * q + 1];
            v4.z = res[4 * q + 2]; v4.w = res[4 * q + 3];
            __builtin_nontemporal_store(v4, (f32x4*)orow + q);
        }
    }
}

extern "C" void kernel_launch(void* const* d_in, const int* in_sizes, int n_in,
                              void* d_out, int out_size, void* d_ws, size_t ws_size,
                              hipStream_t stream) {
    const float* x = (const float*)d_in[0];   // (16384, 4096) f32
    const float* Y = (const float*)d_in[1];   // (2048, 2, 64, 64) f32
    float* out = (float*)d_out;               // (16384, 4096) f32

    const size_t shmem = (size_t)NPAIR * 2 * GG * GG * sizeof(float);  // 256 KB
    (void)hipFuncSetAttribute((const void*)pair_bilinear_tdm_kernel,
                              hipFuncAttributeMaxDynamicSharedMemorySize,
                              (int)shmem);

    dim3 grid(BATCH / BPB, NPAIRS / NPAIR);   // (8, 256)
    pair_bilinear_tdm_kernel<<<grid, NTHREADS, shmem, stream>>>(x, Y, out);
}